// HARP_9577777070302
// MI455X (gfx1250) — compile-verified
//
#include <hip/hip_runtime.h>

#define NB 64
#define NN 10000
#define NE 640000
#define SLOPE 0.02f

typedef __attribute__((ext_vector_type(2))) float v2f;
typedef __attribute__((ext_vector_type(4))) float v4f;
typedef __attribute__((ext_vector_type(8))) float v8f;

// ---------------------------------------------------------------- degree pass
__global__ __launch_bounds__(256) void deg_scatter_k(const int* __restrict__ dst,
                                                     const float* __restrict__ caps,
                                                     float* __restrict__ deg) {
  int e = blockIdx.x * blockDim.x + threadIdx.x;
  int b = blockIdx.y;
  float c = caps[(size_t)b * NE + e];
  atomicAdd(&deg[b * NN + dst[e]], c);
}

__global__ __launch_bounds__(256) void dinv_k(float* __restrict__ d) {
  int i = blockIdx.x * blockDim.x + threadIdx.x;
  d[i] = rsqrtf(d[i] + 1.0f);
}

__global__ __launch_bounds__(256) void init_h_k(const float* __restrict__ x,
                                                float* __restrict__ hbuf) {
  int i = blockIdx.x * blockDim.x + threadIdx.x;
  v2f v = reinterpret_cast<const v2f*>(x)[i];
  v4f h = {v.x, v.y, 0.f, 0.f};
  reinterpret_cast<v4f*>(hbuf)[i] = h;
}

// -------------------------------------------------- xw = h @ W via WMMA f32
// A (16x4 f32): lanes 0-15 hold rows M=0..15 for K=0(V0),K=1(V1);
//               lanes 16-31 hold same rows for K=2(V0),K=3(V1).
// B (4x16 f32): V0 lanes0-15 = row K=0, lanes16-31 = K=2; V1 = K=1 / K=3.
// C (16x16 f32, 8 VGPRs): VGPR r -> M=r (lanes 0-15) / M=8+r (lanes 16-31), N=lane%16.
template <int FIN, int FOUT>
__global__ __launch_bounds__(256) void xw_wmma_k(const float* __restrict__ hbuf,
                                                 const float* __restrict__ W,
                                                 float* __restrict__ xw) {
  const int tilesPer = NN / 16;  // 625
  int wave = blockIdx.x * (blockDim.x >> 5) + (threadIdx.x >> 5);
  int lane = threadIdx.x & 31;
  int b = wave / tilesPer;
  int tbase = (wave - b * tilesPer) * 16;
  int m  = lane & 15;
  int hi = lane >> 4;          // 0 -> K=0,1 ; 1 -> K=2,3
  int kb = hi * 2;

  v4f hv = reinterpret_cast<const v4f*>(hbuf)[b * NN + tbase + m];
  v2f a;
  a.x = hi ? hv.z : hv.x;
  a.y = hi ? hv.w : hv.y;

  v2f bm;
  bm.x = (m < FOUT && kb     < FIN) ? W[kb * FOUT + m]       : 0.f;
  bm.y = (m < FOUT && kb + 1 < FIN) ? W[(kb + 1) * FOUT + m] : 0.f;

  v8f c = {};
  c = __builtin_amdgcn_wmma_f32_16x16x4_f32(false, a, false, bm, (short)0, c,
                                            false, false);

  if (m < 4) {  // only 4 feature columns are meaningful (zero-padded beyond FOUT)
    int base = b * NN + tbase + hi * 8;
#pragma unroll
    for (int r = 0; r < 8; ++r)
      xw[(size_t)(base + r) * 4 + m] = c[r];
  }
}

// ----------------------------------------------------------- edge aggregation
template <int FOUT>
__global__ __launch_bounds__(256) void edge_scatter_k(const int* __restrict__ src,
                                                      const int* __restrict__ dst,
                                                      const float* __restrict__ caps,
                                                      const float* __restrict__ dinv,
                                                      const float* __restrict__ xw,
                                                      float* __restrict__ agg) {
  int e = blockIdx.x * blockDim.x + threadIdx.x;
  int b = blockIdx.y;
  int s = src[e], d = dst[e];
  float cap = caps[(size_t)b * NE + e];
  float nrm = dinv[b * NN + s] * cap * dinv[b * NN + d];
  v4f xv = reinterpret_cast<const v4f*>(xw)[b * NN + s];
  float* ab = agg + (size_t)(b * NN + d) * 4;
  atomicAdd(ab + 0, nrm * xv.x);
  atomicAdd(ab + 1, nrm * xv.y);
  atomicAdd(ab + 2, nrm * xv.z);
  if (FOUT > 3) atomicAdd(ab + 3, nrm * xv.w);
}

template <int FOUT, int COFF>
__global__ __launch_bounds__(256) void finalize_k(const float* __restrict__ agg,
                                                  const float* __restrict__ xw,
                                                  const float* __restrict__ dinv,
                                                  const float* __restrict__ bias,
                                                  float* __restrict__ hbuf,
                                                  float* __restrict__ ne) {
  int i = blockIdx.x * blockDim.x + threadIdx.x;
  float di = dinv[i];
  float sl = di * di;  // self-loop norm
  v4f av = reinterpret_cast<const v4f*>(agg)[i];
  v4f xv = reinterpret_cast<const v4f*>(xw)[i];
  float a[4]  = {av.x, av.y, av.z, av.w};
  float xx[4] = {xv.x, xv.y, xv.z, xv.w};
  float h[4]  = {0.f, 0.f, 0.f, 0.f};
#pragma unroll
  for (int f = 0; f < FOUT; ++f) {
    float v = a[f] + sl * xx[f] + bias[f];
    h[f] = (v >= 0.f) ? v : SLOPE * v;
    ne[(size_t)i * 12 + COFF + f] = h[f];
  }
  v4f hn = {h[0], h[1], h[2], h[3]};
  reinterpret_cast<v4f*>(hbuf)[i] = hn;
}

// ------------------------------------------------------------- edge embedding
__global__ __launch_bounds__(256) void edge_out_k(const int* __restrict__ src,
                                                  const int* __restrict__ dst,
                                                  const float* __restrict__ caps,
                                                  const float* __restrict__ ne,
                                                  float* __restrict__ out) {
  int e = blockIdx.x * blockDim.x + threadIdx.x;
  int b = blockIdx.y;
  __builtin_prefetch(src + e + 8192, 0, 0);   // global_prefetch_b8
  __builtin_prefetch(dst + e + 8192, 0, 0);
  int s = src[e], d = dst[e];
  float cap = caps[(size_t)b * NE + e];
  const v4f* n4 = reinterpret_cast<const v4f*>(ne);
  size_t is = (size_t)(b * NN + s) * 3;
  size_t id = (size_t)(b * NN + d) * 3;
  v4f s0 = n4[is + 0], s1 = n4[is + 1], s2 = n4[is + 2];
  v4f d0 = n4[id + 0], d1 = n4[id + 1], d2 = n4[id + 2];
  v4f o0 = s0 + d0;
  v4f o1 = s1 + d1;
  v4f o2 = s2 + d2;
  o2.w = cap;  // 12th column = capacity (garbage lane 11 of ne overwritten)
  v4f* op = reinterpret_cast<v4f*>(out) + ((size_t)b * NE + e) * 3;
  __builtin_nontemporal_store(o0, op + 0);  // streaming 1.97GB: keep out of L2
  __builtin_nontemporal_store(o1, op + 1);
  __builtin_nontemporal_store(o2, op + 2);
}

// --------------------------------------------------------------------- launch
extern "C" void kernel_launch(void* const* d_in, const int* in_sizes, int n_in,
                              void* d_out, int out_size, void* d_ws, size_t ws_size,
                              hipStream_t stream) {
  (void)in_sizes; (void)n_in; (void)out_size; (void)ws_size;
  const float* x    = (const float*)d_in[0];
  const int*   ei   = (const int*)d_in[1];
  const float* caps = (const float*)d_in[2];
  const float* W0 = (const float*)d_in[3]; const float* b0 = (const float*)d_in[4];
  const float* W1 = (const float*)d_in[5]; const float* b1 = (const float*)d_in[6];
  const float* W2 = (const float*)d_in[7]; const float* b2 = (const float*)d_in[8];
  const int* src = ei;
  const int* dst = ei + NE;

  float* ws   = (float*)d_ws;
  float* deg  = ws;                          // B*N        (becomes dinv in place)
  float* hbuf = deg  + (size_t)NB * NN;      // B*N*4 (layer input, zero padded)
  float* xw   = hbuf + (size_t)NB * NN * 4;  // B*N*4
  float* agg  = xw   + (size_t)NB * NN * 4;  // B*N*4
  float* ne   = agg  + (size_t)NB * NN * 4;  // B*N*12 (cols 0..10 used)

  dim3 tpb(256);
  dim3 gE(NE / 256, NB);               // 2500 x 64, exact
  int  gN = (NB * NN) / 256;           // 2500, exact
  int  gW = (NB * (NN / 16)) / 8;      // 5000 blocks of 8 waves, exact

  hipMemsetAsync(deg, 0, (size_t)NB * NN * sizeof(float), stream);
  deg_scatter_k<<<gE, tpb, 0, stream>>>(dst, caps, deg);
  dinv_k<<<gN, tpb, 0, stream>>>(deg);
  init_h_k<<<gN, tpb, 0, stream>>>(x, hbuf);

  // layer 0: 2 -> 3, ne cols 0..2
  xw_wmma_k<2, 3><<<gW, tpb, 0, stream>>>(hbuf, W0, xw);
  hipMemsetAsync(agg, 0, (size_t)NB * NN * 4 * sizeof(float), stream);
  edge_scatter_k<3><<<gE, tpb, 0, stream>>>(src, dst, caps, deg, xw, agg);
  finalize_k<3, 0><<<gN, tpb, 0, stream>>>(agg, xw, deg, b0, hbuf, ne);

  // layer 1: 3 -> 4, ne cols 3..6
  xw_wmma_k<3, 4><<<gW, tpb, 0, stream>>>(hbuf, W1, xw);
  hipMemsetAsync(agg, 0, (size_t)NB * NN * 4 * sizeof(float), stream);
  edge_scatter_k<4><<<gE, tpb, 0, stream>>>(src, dst, caps, deg, xw, agg);
  finalize_k<4, 3><<<gN, tpb, 0, stream>>>(agg, xw, deg, b1, hbuf, ne);

  // layer 2: 4 -> 4, ne cols 7..10
  xw_wmma_k<4, 4><<<gW, tpb, 0, stream>>>(hbuf, W2, xw);
  hipMemsetAsync(agg, 0, (size_t)NB * NN * 4 * sizeof(float), stream);
  edge_scatter_k<4><<<gE, tpb, 0, stream>>>(src, dst, caps, deg, xw, agg);
  finalize_k<4, 7><<<gN, tpb, 0, stream>>>(agg, xw, deg, b2, hbuf, ne);

  edge_out_k<<<gE, tpb, 0, stream>>>(src, dst, caps, ne, (float*)d_out);
}